// DataTermLayer_53309134078370
// MI455X (gfx1250) — compile-verified
//
#include <hip/hip_runtime.h>

#define ALPHA_F 0.05f

typedef float v4f __attribute__((ext_vector_type(4)));

// Fixed problem geometry from the reference: B=32, H=W=512, C=1.
static constexpr int W = 512;
static constexpr int H = 512;
static constexpr int WM1 = W - 1;
static constexpr int HM1 = H - 1;

// Each thread handles a quad of 4 consecutive x-pixels:
//   flow  : 2x v4f NT loads     (read-once stream -> TH_NT, keep L2 for I1/I2)
//   I1    : v4f (+1 scalar) row y, v4f row y+1 (cached, reused by neighbors)
//   I2    : 16 scalar bilinear gather taps (cached; flow ~N(0,1) => high locality)
//   out   : 2x v4f NT stores    (write-once stream -> TH_NT)
// All edge guards are branchless (clamped addresses + zeroed gradients), so
// every load is unconditional and clause-friendly.
__global__ __launch_bounds__(256)
void dataterm_kernel(const float* __restrict__ I1,
                     const float* __restrict__ I2,
                     const float* __restrict__ flow,
                     float* __restrict__ out,
                     int nQuads)
{
    int quad = blockIdx.x * blockDim.x + threadIdx.x;
    if (quad >= nQuads) return;

    const int xq  = quad & (W / 4 - 1);   // quad index within row
    const int row = quad >> 7;            // global row = b*H + y   (W/4 = 128)
    const int y   = row & (H - 1);
    const int b   = row >> 9;             // H = 512

    const int  x = xq << 2;
    const long p = (long)row * W + x;     // flat pixel index into (B*H*W)

    // ---- flow (u,v interleaved): 8 floats, 32B-aligned, NT ----
    const v4f* fptr = (const v4f*)(flow + p * 2);
    const v4f f01 = __builtin_nontemporal_load(fptr);
    const v4f f23 = __builtin_nontemporal_load(fptr + 1);
    const float u[4] = { f01.x, f01.z, f23.x, f23.z };
    const float v[4] = { f01.y, f01.w, f23.y, f23.w };

    // ---- I1 center row: x..x+3 plus one extra for the horizontal diff ----
    // Branchless edge handling: at the last quad of a row, re-load c[3]'s
    // address; the value is unused because dx is zeroed at x = W-1.
    const v4f c4 = *(const v4f*)(I1 + p);
    const float cExtra = I1[p + ((x + 4 < W) ? 4 : 3)];
    const float c[5] = { c4.x, c4.y, c4.z, c4.w, cExtra };

    // ---- I1 next row (vertical diff) ----
    // Branchless: at the last row, reload the current row; gy is zeroed below.
    const bool hasDown = (y < HM1);
    const v4f d4 = *(const v4f*)(I1 + p + (hasDown ? W : 0));
    const float dn[4] = { d4.x, d4.y, d4.z, d4.w };

    const float* __restrict__ I2b = I2 + (long)b * (H * W);

    float res[8];
#pragma unroll
    for (int i = 0; i < 4; ++i) {
        const int xi = x + i;

        // clamped sample coordinates
        const float fx = fminf(fmaxf((float)xi + u[i], 0.0f), (float)WM1);
        const float fy = fminf(fmaxf((float)y  + v[i], 0.0f), (float)HM1);
        const float x0f = floorf(fx);
        const float y0f = floorf(fy);
        const int x0 = (int)x0f;
        const int y0 = (int)y0f;
        const int x1 = min(x0 + 1, WM1);
        const int y1 = min(y0 + 1, HM1);
        const float wx = fx - x0f;
        const float wy = fy - y0f;

        // 4 bilinear taps (RT-cached; heavy spatial reuse across lanes/taps)
        const float* __restrict__ r0 = I2b + y0 * W;
        const float* __restrict__ r1 = I2b + y1 * W;
        const float Ia = r0[x0];
        const float Ib = r0[x1];
        const float Ic = r1[x0];
        const float Id = r1[x1];

        // exact reference formula: (Ia(1-wx)+Ib wx)(1-wy) + (Ic(1-wx)+Id wx)wy
        const float top    = Ia * (1.0f - wx) + Ib * wx;
        const float bot    = Ic * (1.0f - wx) + Id * wx;
        const float warped = top * (1.0f - wy) + bot * wy;

        const float dataTerm = warped - c[i];

        // NOTE: reference swaps names: grad_x = dy (vertical), grad_y = dx (horizontal)
        const float gy = hasDown    ? (dn[i]    - c[i]) : 0.0f;  // dy -> updates u
        const float gx = (xi < WM1) ? (c[i + 1] - c[i]) : 0.0f;  // dx -> updates v

        res[2 * i + 0] = u[i] - ALPHA_F * dataTerm * gy;
        res[2 * i + 1] = v[i] - ALPHA_F * dataTerm * gx;
    }

    // ---- output (u,v interleaved): 2x v4f NT stores, 32B-aligned ----
    v4f* optr = (v4f*)(out + p * 2);
    const v4f o01 = (v4f){res[0], res[1], res[2], res[3]};
    const v4f o23 = (v4f){res[4], res[5], res[6], res[7]};
    __builtin_nontemporal_store(o01, optr);
    __builtin_nontemporal_store(o23, optr + 1);
}

extern "C" void kernel_launch(void* const* d_in, const int* in_sizes, int n_in,
                              void* d_out, int out_size, void* d_ws, size_t ws_size,
                              hipStream_t stream)
{
    const float* I1   = (const float*)d_in[0];   // (B,H,W,1) f32
    const float* I2   = (const float*)d_in[1];   // (B,H,W,1) f32
    const float* flow = (const float*)d_in[2];   // (B,H,W,2) f32
    float*       out  = (float*)d_out;           // (B,H,W,2) f32

    const int nPixels = in_sizes[0];             // B*H*W
    const int nQuads  = nPixels >> 2;            // 4 px per thread

    const int block = 256;                       // 8 wave32s per block
    const int grid  = (nQuads + block - 1) / block;

    dataterm_kernel<<<grid, block, 0, stream>>>(I1, I2, flow, out, nQuads);
}